// TMRPCEN_1322849927827
// MI455X (gfx1250) — compile-verified
//
#include <hip/hip_runtime.h>

typedef float v2f __attribute__((ext_vector_type(2)));
typedef float v4f __attribute__((ext_vector_type(4)));
typedef float v8f __attribute__((ext_vector_type(8)));

#define EPS     1e-5f
#define NR      10
#define NB      16
#define NF      128
#define NT      2048
#define LOG2E   1.4426950408889634f

// Raw hardware transcendentals (no OCML denorm/inf guard code):
//   v_log_f32 computes log2(x), v_exp_f32 computes 2^x.
__device__ __forceinline__ float hw_log2(float x) { return __builtin_amdgcn_logf(x); }
__device__ __forceinline__ float hw_exp2(float x) { return __builtin_amdgcn_exp2f(x); }
__device__ __forceinline__ float hw_exp(float x)  { return hw_exp2(x * LOG2E); }

// One wave32 handles one (rate r, freq f) pair for ALL 16 batches and all T.
// Within a 16-step time chunk the EMA scan is a 16x16 matmul:
//   Y = S @ X + d * carry,  S[i,k] = s*(1-s)^(i-k) (k<=i), d[i] = (1-s)^(i+1)
// computed as 4 chained V_WMMA_F32_16X16X4_F32 (exact fp32).
// C/D layout (CDNA5 ISA): lane = n + 16*(t>=8), vgpr v = t%8.
// A 16x4 layout: lane m=0..15 holds M=m; VGPR v holds K = 2*(lane>=16)+v.
// B 4x16 layout (mirror): lane holds N=lane%16; VGPR v holds K = 2*(lane>=16)+v.
__global__ __launch_bounds__(128) void pcen_wmma_scan(
    const float* __restrict__ x,          // (B, F, T)
    const float* __restrict__ s_log,      // (R, F)
    const float* __restrict__ alpha_log,  // (F)
    const float* __restrict__ delta_log,  // (F)
    const float* __restrict__ r_log,      // (F)
    float* __restrict__ out)              // (B, R, F, T)
{
    const int lane  = threadIdx.x & 31;
    const int wave  = (blockIdx.x * (int)(blockDim.x >> 5)) + (threadIdx.x >> 5);
    if (wave >= NR * NF) return;
    const int r = wave / NF;
    const int f = wave % NF;

    const int m    = lane & 15;   // A-row / B-col / batch index
    const int half = lane >> 4;   // lane group 0/1

    // wave-uniform per-(r,f) parameters (all in base-2 form where useful)
    const float s        = hw_exp(s_log[r * NF + f]);
    const float alpha    = hw_exp(alpha_log[f]);
    const float delta    = hw_exp(delta_log[f]);
    const float rr       = hw_exp(r_log[f]);
    const float delta_r  = hw_exp2(rr * hw_log2(delta));
    const float oms      = 1.0f - s;
    const float invEPS   = 1.0f / EPS;
    const float negAlpha = -alpha;
    // c0 = -alpha * log2(EPS): folds the eps term into a single fma later
    const float c0       = negAlpha * hw_log2(EPS);

    // A-matrix fragments (constant over the whole T loop).
    // (1-s)^p via exact repeated multiply to avoid log/exp error in the scan.
    v2f a[4];
    {
        float pw[16];   // pw[p] = (1-s)^p
        pw[0] = 1.0f;
#pragma unroll
        for (int p = 1; p < 16; ++p) pw[p] = pw[p - 1] * oms;
#pragma unroll
        for (int j = 0; j < 4; ++j) {
#pragma unroll
            for (int v = 0; v < 2; ++v) {
                const int k = 4 * j + 2 * half + v;
                a[j][v] = (k <= m) ? s * pw[m - k] : 0.0f;
            }
        }
    }
    // carry-injection vector for this lane's 8 C rows (t = v + 8*half):
    // dvec[v] = (1-s)^(t+1)
    float dvec[8];
    {
        float base = oms;                       // (1-s)^1
#pragma unroll
        for (int h = 0; h < 8 * 1; ++h) { }     // no-op keep structure
        if (half) {
            // (1-s)^9 start for lanes 16..31
#pragma unroll
            for (int p = 0; p < 8; ++p) base *= oms;
        }
        float cur = base;
#pragma unroll
        for (int v = 0; v < 8; ++v) { dvec[v] = cur; cur *= oms; }
    }

    // this lane's input channel (batch = m) and output channel
    const float* xc = x   + ((size_t)m * NF + f) * NT;
    float*       oc = out + (((size_t)m * NR + r) * NF + f) * NT;

    // y[0] = x[0] exactly <=> virtual carry y[-1] = x[0]
    float carry = xc[0];

    for (int t0 = 0; t0 < NT; t0 += 16) {
        // C init: outer product d * carry (all lanes hold carry of channel m)
        v8f acc;
#pragma unroll
        for (int v = 0; v < 8; ++v) acc[v] = dvec[v] * carry;

        // 4 chained K=4 WMMAs cover the 16x16 intra-chunk scan matrix
#pragma unroll
        for (int j = 0; j < 4; ++j) {
            const v2f bfrag = *(const v2f*)(xc + t0 + 4 * j + 2 * half);
            acc = __builtin_amdgcn_wmma_f32_16x16x4_f32(
                false, a[j], false, bfrag, (short)0, acc, false, false);
        }

        // new carry = y[t0+15] of channel m: C VGPR 7, lane m+16
        carry = __shfl(acc[7], 16 + m, 32);

        // elementwise PCEN on the 8 smoother values this lane owns.
        // smooth = (EPS + y)^(-alpha) = exp2(-alpha*log2(1 + y/EPS) + c0)
        // pcen   = exp2(rr*log2(x*smooth + delta)) - delta^rr
        const v4f x0 = *(const v4f*)(xc + t0 + 8 * half);
        const v4f x1 = *(const v4f*)(xc + t0 + 8 * half + 4);
        v4f o0, o1;
#pragma unroll
        for (int v = 0; v < 8; ++v) {
            const float y    = acc[v];
            const float xv   = (v < 4) ? x0[v] : x1[v - 4];
            const float lg   = hw_log2(__builtin_fmaf(y, invEPS, 1.0f));
            const float sm   = hw_exp2(__builtin_fmaf(negAlpha, lg, c0));
            const float base = __builtin_fmaf(xv, sm, delta);
            const float p    = hw_exp2(rr * hw_log2(base)) - delta_r;
            if (v < 4) o0[v] = p; else o1[v - 4] = p;
        }
        *(v4f*)(oc + t0 + 8 * half)     = o0;
        *(v4f*)(oc + t0 + 8 * half + 4) = o1;
    }
}

extern "C" void kernel_launch(void* const* d_in, const int* in_sizes, int n_in,
                              void* d_out, int out_size, void* d_ws, size_t ws_size,
                              hipStream_t stream) {
    const float* x         = (const float*)d_in[0];
    const float* s_log     = (const float*)d_in[1];
    const float* alpha_log = (const float*)d_in[2];
    const float* delta_log = (const float*)d_in[3];
    const float* r_log     = (const float*)d_in[4];
    float*       out       = (float*)d_out;

    // 10*128 = 1280 waves, 4 waves (128 threads) per block -> 320 blocks
    const int wavesTotal = NR * NF;
    const int wavesPerBlock = 4;
    const int blocks = (wavesTotal + wavesPerBlock - 1) / wavesPerBlock;
    pcen_wmma_scan<<<blocks, wavesPerBlock * 32, 0, stream>>>(
        x, s_log, alpha_log, delta_log, r_log, out);
}